// EMAVectorQuantizer_58471684768003
// MI455X (gfx1250) — compile-verified
//
#include <hip/hip_runtime.h>
#include <hip/hip_bf16.h>
#include <cstdint>

// ---------------------------------------------------------------------------
// EMA Vector Quantizer for MI455X (gfx1250, wave32)
//   z:            (16, 64, 32, 32) f32   -> T = 16384 tokens, d = 64
//   weight:       (8192, 64) f32
//   cluster_size: (8192,) f32
//   embed_avg:    (8192, 64) f32
// Dominant work: 16384x8192x64 GEMM + argmin -> fp32 WMMA (16x16x4).
// ---------------------------------------------------------------------------

#define T_TOK    16384
#define NCODE    8192
#define DIM      64
#define CHUNK    128          // codes staged per LDS round
#define LDSTRIDE 68           // dwords per code row in LDS (pad: banks + 16B align)

typedef float v2f __attribute__((ext_vector_type(2)));
typedef float v8f __attribute__((ext_vector_type(8)));

// ---------------------------------------------------------------------------
// Kernel 0: wsq[n] = ||weight[n]||^2 ; zero counts / embed_sum / scalars
// ---------------------------------------------------------------------------
__global__ __launch_bounds__(256)
void vq_prep_kernel(const float* __restrict__ weight,
                    float* __restrict__ wsq,
                    float* __restrict__ counts,
                    float* __restrict__ embed_sum,
                    float* __restrict__ scal) {
    int i = blockIdx.x * 256 + threadIdx.x;           // grid covers 524288
    if (i < NCODE * DIM) embed_sum[i] = 0.0f;
    if (i < NCODE) {
        counts[i] = 0.0f;
        const float* wr = weight + (size_t)i * DIM;
        float s = 0.0f;
        #pragma unroll
        for (int c = 0; c < DIM; ++c) s = fmaf(wr[c], wr[c], s);
        wsq[i] = s;
    }
    if (i < 4) scal[i] = 0.0f;
}

// ---------------------------------------------------------------------------
// Kernel 1: WMMA distance + argmin.
//   1 wave  = one 16-token tile (A kept in VGPRs, 16 x v2f frags for K=64)
//   1 block = 8 waves = 128 tokens; loops over all 8192 codes in 128-code
//             LDS-staged chunks (async-to-LDS), 16-code WMMA tiles inside.
//   dist-equivalent value: ||w_n||^2 - 2 z.w  (||z||^2 is constant per row,
//   dropped -- does not change argmin).
// ---------------------------------------------------------------------------
__global__ __launch_bounds__(256)
void vq_argmin_kernel(const float* __restrict__ z,
                      const float* __restrict__ weight,
                      const float* __restrict__ wsq,
                      int*   __restrict__ idx_i,
                      float* __restrict__ idx_f) {
    __shared__ float sw[CHUNK * LDSTRIDE];   // staged codebook chunk
    __shared__ float sq[CHUNK];              // staged ||w||^2 chunk

    const int tid  = threadIdx.x;
    const int wave = tid >> 5;
    const int lane = tid & 31;
    const int half = lane >> 4;              // 0: lanes 0-15, 1: lanes 16-31
    const int lm   = lane & 15;

    // ---- load this wave's A tile: tokens t0..t0+15, all 64 dims -----------
    // A-matrix 16x4 f32 layout: lane -> M = lane&15 ; VGPR v, half h -> K = 2h+v
    const int t0    = (blockIdx.x * 8 + wave) * 16;
    const int token = t0 + lm;
    const int bimg  = token >> 10;           // 1024 tokens (32x32) per image
    const int hw    = token & 1023;
    const float* zb = z + (size_t)bimg * 65536 + hw;   // stride over c = 1024

    v2f a[16];
    #pragma unroll
    for (int q = 0; q < 16; ++q) {
        const int k = q * 4 + half * 2;
        a[q].x = zb[(size_t)k * 1024];
        a[q].y = zb[(size_t)(k + 1) * 1024];
    }

    float minv[8];
    int   mini[8];
    #pragma unroll
    for (int r = 0; r < 8; ++r) { minv[r] = 3.0e38f; mini[r] = 0; }

    for (int chunk = 0; chunk < NCODE / CHUNK; ++chunk) {
        // ---- async-stage 128 codes (8192 floats) into LDS as b128 ---------
        const float* wc = weight + (size_t)chunk * CHUNK * DIM;
        #pragma unroll
        for (int i = 0; i < 8; ++i) {
            const int g    = i * 256 + tid;            // 16B group id, 2048 total
            const int row  = g >> 4;                   // 16 groups per 64-dword row
            const int c4   = (g & 15) * 4;
            unsigned ldsb  = (unsigned)(size_t)&sw[row * LDSTRIDE + c4]; // byte addr
            unsigned voff  = (unsigned)(g * 16);       // byte offset from wc
            asm volatile("global_load_async_to_lds_b128 %0, %1, %2 offset:0"
                         :: "v"(ldsb), "v"(voff), "s"(wc)
                         : "memory");
        }
        if (tid < CHUNK) sq[tid] = wsq[chunk * CHUNK + tid];
        asm volatile("s_wait_asynccnt 0" ::: "memory");
        __syncthreads();

        // ---- 8 WMMA tiles of 16 codes each --------------------------------
        #pragma unroll 1
        for (int ct = 0; ct < CHUNK / 16; ++ct) {
            // B 4x16 f32 frag: lane -> N = lane&15 ; VGPR v, half h -> K = 2h+v
            const float* srow = &sw[(ct * 16 + lm) * LDSTRIDE];
            v2f bf[16];
            #pragma unroll
            for (int q = 0; q < 16; ++q) {
                const int k = q * 4 + half * 2;
                bf[q].x = srow[k];
                bf[q].y = srow[k + 1];
            }
            v8f acc = {0.f, 0.f, 0.f, 0.f, 0.f, 0.f, 0.f, 0.f};
            #pragma unroll
            for (int q = 0; q < 16; ++q)
                acc = __builtin_amdgcn_wmma_f32_16x16x4_f32(
                          false, a[q], false, bf[q],
                          (short)0, acc, false, false);

            const int   n  = chunk * CHUNK + ct * 16 + lm;   // same N for all 8 rows
            const float wq = sq[ct * 16 + lm];
            #pragma unroll
            for (int r = 0; r < 8; ++r) {                    // row M = r + 8*half
                const float val = fmaf(-2.0f, acc[r], wq);
                if (val < minv[r]) { minv[r] = val; mini[r] = n; }
            }
        }
        __syncthreads();
    }

    // ---- argmin across the 16 lanes sharing each M row (wave32 halves) ----
    #pragma unroll
    for (int r = 0; r < 8; ++r) {
        #pragma unroll
        for (int off = 8; off > 0; off >>= 1) {
            const float ov = __shfl_xor(minv[r], off, 32);
            const int   oi = __shfl_xor(mini[r], off, 32);
            if (ov < minv[r] || (ov == minv[r] && oi < mini[r])) {
                minv[r] = ov; mini[r] = oi;
            }
        }
    }
    if (lm == 0) {                       // lane 0 -> rows 0..7, lane 16 -> rows 8..15
        #pragma unroll
        for (int r = 0; r < 8; ++r) {
            const int t = t0 + half * 8 + r;
            idx_i[t] = mini[r];
            idx_f[t] = (float)mini[r];
        }
    }
}

// ---------------------------------------------------------------------------
// Kernel 2: gather z_q, straight-through output (b,c,h,w), loss partials,
//           counts and embed_sum segment sums (float atomics).
//   e: c = e>>14 (0..63), t = e&16383 -> coalesced z reads & out writes.
// ---------------------------------------------------------------------------
__global__ __launch_bounds__(256)
void vq_scatter_kernel(const float* __restrict__ z,
                       const float* __restrict__ weight,
                       const int*   __restrict__ idx_i,
                       float* __restrict__ out,
                       float* __restrict__ embed_sum,
                       float* __restrict__ counts,
                       float* __restrict__ scal) {
    const int e = blockIdx.x * 256 + threadIdx.x;   // 1048576 total
    const int c = e >> 14;
    const int t = e & (T_TOK - 1);
    const int id = idx_i[t];
    const float wv = weight[(size_t)id * DIM + c];
    const int b  = t >> 10;
    const int hw = t & 1023;
    const size_t zi = (size_t)b * 65536 + (size_t)c * 1024 + hw;
    const float zv = z[zi];
    const float diff = wv - zv;
    out[zi] = zv + diff;                            // zc + (z_q - zc), bit-matched

    atomicAdd(&embed_sum[(size_t)id * DIM + c], zv);
    if (c == 0) atomicAdd(&counts[id], 1.0f);

    float d2 = diff * diff;                         // wave32 reduce, then atomic
    #pragma unroll
    for (int off = 16; off > 0; off >>= 1) d2 += __shfl_xor(d2, off, 32);
    if ((threadIdx.x & 31) == 0) atomicAdd(&scal[0], d2);
}

// ---------------------------------------------------------------------------
// Kernel 3: EMA updates + scalar sums (n, perplexity accumulator).
// ---------------------------------------------------------------------------
__global__ __launch_bounds__(256)
void vq_ema_kernel(const float* __restrict__ cluster_size,
                   const float* __restrict__ embed_avg,
                   const float* __restrict__ counts,
                   const float* __restrict__ embed_sum,
                   float* __restrict__ out_cs,
                   float* __restrict__ out_ea,
                   float* __restrict__ scal) {
    const int e = blockIdx.x * 256 + threadIdx.x;   // 524288 total
    const int n = e >> 6;
    const int c = e & (DIM - 1);
    const float na = fmaf(0.99f, embed_avg[e], 0.01f * embed_sum[e]);
    out_ea[e] = na;
    if (c == 0) {
        const float cnt = counts[n];
        const float ncs = fmaf(0.99f, cluster_size[n], 0.01f * cnt);
        out_cs[n] = ncs;
        atomicAdd(&scal[1], ncs);                           // n = sum(new_cs)
        const float p = cnt * (1.0f / (float)T_TOK);
        atomicAdd(&scal[2], p * logf(p + 1e-10f));          // sum p*log(p+eps)
    }
}

// ---------------------------------------------------------------------------
// Kernel 4: normalized codebook + loss & perplexity scalars.
// ---------------------------------------------------------------------------
__global__ __launch_bounds__(256)
void vq_finalize_kernel(const float* __restrict__ out_cs,
                        const float* __restrict__ out_ea,
                        float* __restrict__ out_w,
                        const float* __restrict__ scal,
                        float* __restrict__ loss_out,
                        float* __restrict__ perp_out) {
    const int e = blockIdx.x * 256 + threadIdx.x;   // 524288 total
    const int n = e >> 6;
    const float nsum = scal[1];
    const float smoothed = (out_cs[n] + 1e-5f) / (nsum + (float)NCODE * 1e-5f) * nsum;
    out_w[e] = out_ea[e] / smoothed;
    if (e == 0) {
        loss_out[0] = 0.25f * scal[0] * (1.0f / 1048576.0f);
        perp_out[0] = expf(-scal[2]);
    }
}

// ---------------------------------------------------------------------------
// Launcher
// ---------------------------------------------------------------------------
extern "C" void kernel_launch(void* const* d_in, const int* in_sizes, int n_in,
                              void* d_out, int out_size, void* d_ws, size_t ws_size,
                              hipStream_t stream) {
    const float* z            = (const float*)d_in[0];   // 16*64*32*32
    const float* weight       = (const float*)d_in[1];   // 8192*64
    const float* cluster_size = (const float*)d_in[2];   // 8192
    const float* embed_avg    = (const float*)d_in[3];   // 8192*64

    float* out = (float*)d_out;
    // output tuple layout (flat, return order):
    float* o_out  = out;                      // 1048576
    float* o_loss = out + 1048576;            // 1
    float* o_perp = out + 1048577;            // 1
    float* o_idx  = out + 1048578;            // 16384 (written as float values)
    float* o_cs   = out + 1064962;            // 8192
    float* o_ea   = out + 1073154;            // 524288
    float* o_w    = out + 1597442;            // 524288

    // workspace layout
    float* ws        = (float*)d_ws;
    float* wsq       = ws;                    // 8192
    float* counts    = ws + 8192;             // 8192
    float* embed_sum = ws + 16384;            // 524288
    float* scal      = ws + 540672;           // 4: [loss_sum, n_sum, perp_sum, -]
    int*   idx_i     = (int*)(ws + 540676);   // 16384

    hipLaunchKernelGGL(vq_prep_kernel, dim3(2048), dim3(256), 0, stream,
                       weight, wsq, counts, embed_sum, scal);
    hipLaunchKernelGGL(vq_argmin_kernel, dim3(T_TOK / 128), dim3(256), 0, stream,
                       z, weight, wsq, idx_i, o_idx);
    hipLaunchKernelGGL(vq_scatter_kernel, dim3(4096), dim3(256), 0, stream,
                       z, weight, idx_i, o_out, embed_sum, counts, scal);
    hipLaunchKernelGGL(vq_ema_kernel, dim3(2048), dim3(256), 0, stream,
                       cluster_size, embed_avg, counts, embed_sum, o_cs, o_ea, scal);
    hipLaunchKernelGGL(vq_finalize_kernel, dim3(2048), dim3(256), 0, stream,
                       o_cs, o_ea, o_w, scal, o_loss, o_perp);
}